// SimpleExpectationValueModel_50405736186384
// MI455X (gfx1250) — compile-verified
//
#include <hip/hip_runtime.h>
#include <hip/hip_bf16.h>

// ---------------------------------------------------------------------------
// CDNA5 (gfx1250) implementation of SimpleExpectationValueModel forward pass.
// Big GEMMs: V_WMMA_F32_16X16X4_F32 (exact fp32 matrix pipe), weight panels
// staged into LDS with GLOBAL_LOAD_ASYNC_TO_LDS_B32 (ASYNCcnt path).
// ---------------------------------------------------------------------------

typedef __attribute__((ext_vector_type(2))) float v2f;
typedef __attribute__((ext_vector_type(8))) float v8f;

__device__ __forceinline__ v8f wmma_f32_16x16x4(v2f a, v2f b, v8f c) {
  return __builtin_amdgcn_wmma_f32_16x16x4_f32(
      /*neg_a=*/false, a, /*neg_b=*/false, b,
      /*c_mod=*/(short)0, c, /*reuse_a=*/false, /*reuse_b=*/false);
}

// Async copy one dword global -> LDS (tracked by ASYNCcnt).
// lds_byte: byte offset within the workgroup LDS allocation (low 32 bits of
// the generic shared pointer; hardware truncates flat LDS addrs to [31:0]).
__device__ __forceinline__ void async_g2l_b32(unsigned lds_byte,
                                              const float* gptr) {
  asm volatile("global_load_async_to_lds_b32 %0, %1, off"
               :
               : "v"(lds_byte), "v"(gptr)
               : "memory");
}
__device__ __forceinline__ void wait_async0() {
  asm volatile("s_wait_asynccnt 0x0" ::: "memory");
}

#define NEG_INF (-3.402823466e38f)

// ---------------------------------------------------------------------------
__global__ void zero_u8_kernel(unsigned char* p, size_t n) {
  size_t i = (size_t)blockIdx.x * blockDim.x + threadIdx.x;
  if (i < n) p[i] = 0;
}

// mask1[b,i,j] = 1 iff edge j->i  (edges[0]=src, edges[1]=dst, global ids)
__global__ void edges_kernel(const int* edges, unsigned char* mask1, int E) {
  int e = blockIdx.x * blockDim.x + threadIdx.x;
  if (e >= E) return;
  int s = edges[e];
  int d = edges[E + e];
  mask1[((size_t)(d >> 7)) * 16384 + (size_t)(d & 127) * 128 + (size_t)(s & 127)] = 1;
}

// ---------------------------------------------------------------------------
// C[M,N] = A[M,K] @ W[K,N] + bias[N]   (shared weights)
// grid (M/64, N/16), block 128 = 4 waves, each wave one 16x16 tile.
// Weight panel W[:,n0:n0+16] async-staged into LDS once per block.
// K <= 384, M % 64 == 0, N % 16 == 0.
// ---------------------------------------------------------------------------
__global__ __launch_bounds__(128) void gemm_bias_kernel(
    const float* __restrict__ A, const float* __restrict__ W,
    const float* __restrict__ bias, float* __restrict__ C,
    int M, int N, int K) {
  __shared__ float panel[384 * 16];  // 24 KB max
  int t = threadIdx.x;
  int n0 = blockIdx.y << 4;
  unsigned pbase = (unsigned)(size_t)(void*)panel;
  for (int e = t; e < K * 16; e += 128) {
    int kr = e >> 4, c = e & 15;
    async_g2l_b32(pbase + (unsigned)e * 4u, W + (size_t)kr * N + n0 + c);
  }
  wait_async0();
  __syncthreads();

  int wave = t >> 5, lane = t & 31;
  int lm = lane & 15, hi = lane >> 4;
  int m0 = (blockIdx.x << 6) + (wave << 4);
  if (m0 >= M) return;
  v8f acc = {};
  for (int k = 0; k < K; k += 4) {
    v2f a, b;
    const float* arow = A + (size_t)(m0 + lm) * K + k + 2 * hi;
    a.x = arow[0];
    a.y = arow[1];
    b.x = panel[(k + 2 * hi) * 16 + lm];
    b.y = panel[(k + 2 * hi + 1) * 16 + lm];
    acc = wmma_f32_16x16x4(a, b, acc);
  }
  float bv = bias ? bias[n0 + lm] : 0.f;
#pragma unroll
  for (int r = 0; r < 8; ++r) {
    int m = m0 + r + 8 * hi;
    C[(size_t)m * N + n0 + lm] = acc[r] + bv;
  }
}

// ---------------------------------------------------------------------------
// Batched per-graph GEMM:  C_b[M,N] = A_b[M,K] @ B_b[K,N]
// grid (N/16, nbatch); block 128 = 4 waves; B panel async-staged to LDS;
// each wave sweeps M/64 tiles.  K <= 128, M % 16 == 0.
// ---------------------------------------------------------------------------
__global__ __launch_bounds__(128) void gemm_batched_kernel(
    const float* __restrict__ A, const float* __restrict__ Bm,
    float* __restrict__ C, int M, int N, int K,
    size_t sA, size_t sB, size_t sC) {
  __shared__ float panel[128 * 16];  // 8 KB
  int t = threadIdx.x;
  int n0 = blockIdx.x << 4;
  const float* Ab = A + (size_t)blockIdx.y * sA;
  const float* Bb = Bm + (size_t)blockIdx.y * sB;
  float* Cb = C + (size_t)blockIdx.y * sC;

  unsigned pbase = (unsigned)(size_t)(void*)panel;
  for (int e = t; e < K * 16; e += 128) {
    int kr = e >> 4, c = e & 15;
    async_g2l_b32(pbase + (unsigned)e * 4u, Bb + (size_t)kr * N + n0 + c);
  }
  wait_async0();
  __syncthreads();

  int wave = t >> 5, lane = t & 31;
  int lm = lane & 15, hi = lane >> 4;
  for (int mt = wave; mt < (M >> 4); mt += 4) {
    int m0 = mt << 4;
    v8f acc = {};
    for (int k = 0; k < K; k += 4) {
      v2f a, b;
      const float* arow = Ab + (size_t)(m0 + lm) * K + k + 2 * hi;
      a.x = arow[0];
      a.y = arow[1];
      b.x = panel[(k + 2 * hi) * 16 + lm];
      b.y = panel[(k + 2 * hi + 1) * 16 + lm];
      acc = wmma_f32_16x16x4(a, b, acc);
    }
#pragma unroll
    for (int r = 0; r < 8; ++r)
      Cb[(size_t)(m0 + r + 8 * hi) * N + n0 + lm] = acc[r];
  }
}

// ---------------------------------------------------------------------------
// TransformerConv stage 1 attention: per (head h, graph b) block.
// logits = q k^T / sqrt(128) via WMMA into LDS, masked softmax, out = a@v via
// WMMA, accumulated into x1 (which holds the root_weight skip x@Ws+bs).
// ---------------------------------------------------------------------------
__global__ __launch_bounds__(256) void attn1_kernel(
    const float* __restrict__ q, const float* __restrict__ k,
    const float* __restrict__ v, const unsigned char* __restrict__ mask,
    float* __restrict__ x1) {
  __shared__ float sm[128 * 128];  // 64 KB
  int h = blockIdx.x, b = blockIdx.y;
  int wave = threadIdx.x >> 5, lane = threadIdx.x & 31;
  int lm = lane & 15, hi = lane >> 4;
  const float scale = 0.08838834764831845f;  // 1/sqrt(128)
  size_t base = (size_t)b * 128 * 384 + (size_t)h * 128;

  for (int t = wave; t < 64; t += 8) {
    int m0 = (t >> 3) << 4, n0 = (t & 7) << 4;
    v8f acc = {};
    for (int kk = 0; kk < 128; kk += 4) {
      v2f a, bb;
      const float* qr = q + base + (size_t)(m0 + lm) * 384 + kk + 2 * hi;
      a.x = qr[0];
      a.y = qr[1];
      const float* kr = k + base + (size_t)(n0 + lm) * 384 + kk + 2 * hi;
      bb.x = kr[0];
      bb.y = kr[1];
      acc = wmma_f32_16x16x4(a, bb, acc);
    }
#pragma unroll
    for (int r = 0; r < 8; ++r)
      sm[(m0 + r + 8 * hi) * 128 + n0 + lm] = acc[r] * scale;
  }
  __syncthreads();

  if (threadIdx.x < 128) {
    int i = threadIdx.x;
    const unsigned char* mrow = mask + ((size_t)b * 128 + i) * 128;
    float mx = NEG_INF;
    for (int j = 0; j < 128; ++j)
      if (mrow[j]) mx = fmaxf(mx, sm[i * 128 + j]);
    float s = 0.f;
    for (int j = 0; j < 128; ++j) {
      float e = mrow[j] ? expf(sm[i * 128 + j] - mx) : 0.f;
      sm[i * 128 + j] = e;
      s += e;
    }
    float inv = 1.f / fmaxf(s, 1e-16f);
    for (int j = 0; j < 128; ++j) sm[i * 128 + j] *= inv;
  }
  __syncthreads();

  for (int t = wave; t < 64; t += 8) {
    int m0 = (t >> 3) << 4, n0 = (t & 7) << 4;
    v8f acc = {};
    for (int kk = 0; kk < 128; kk += 4) {
      v2f a, bb;
      a.x = sm[(m0 + lm) * 128 + kk + 2 * hi];
      a.y = sm[(m0 + lm) * 128 + kk + 2 * hi + 1];
      bb.x = v[base + (size_t)(kk + 2 * hi) * 384 + n0 + lm];
      bb.y = v[base + (size_t)(kk + 2 * hi + 1) * 384 + n0 + lm];
      acc = wmma_f32_16x16x4(a, bb, acc);
    }
#pragma unroll
    for (int r = 0; r < 8; ++r) {
      size_t o = (size_t)(b * 128 + m0 + r + 8 * hi) * 384 + h * 128 + n0 + lm;
      x1[o] += acc[r];
    }
  }
}

// ---------------------------------------------------------------------------
// ASAP scatter-max: out[b,i,c] = max_{j in N(i) U {i}} x1[b,j,c]
// ---------------------------------------------------------------------------
__global__ __launch_bounds__(128) void scatmax_kernel(
    const float* __restrict__ x1, const unsigned char* __restrict__ mask,
    float* __restrict__ out) {
  __shared__ unsigned char flg[128];
  int bi = blockIdx.x;  // b*128 + i
  int b = bi >> 7, i = bi & 127;
  int t = threadIdx.x;
  flg[t] = (t == i) ? 1 : mask[((size_t)b * 128 + i) * 128 + t];
  __syncthreads();
  for (int c = t; c < 384; c += 128) {
    float mx = NEG_INF;
    for (int j = 0; j < 128; ++j)
      if (flg[j]) mx = fmaxf(mx, x1[((size_t)b * 128 + j) * 384 + c]);
    out[(size_t)bi * 384 + c] = mx;
  }
}

// y[row] = A[row,:] . w + bias  (one wave per row)
__global__ __launch_bounds__(256) void matvec_kernel(
    const float* __restrict__ A, const float* __restrict__ w,
    const float* __restrict__ bias, float* __restrict__ y, int M, int K) {
  int wave = threadIdx.x >> 5, lane = threadIdx.x & 31;
  int row = blockIdx.x * 8 + wave;
  if (row >= M) return;
  const float* a = A + (size_t)row * K;
  float s = 0.f;
  for (int kk = lane; kk < K; kk += 32) s += a[kk] * w[kk];
  for (int o = 16; o; o >>= 1) s += __shfl_down(s, o, 32);
  if (lane == 0) y[row] = s + (bias ? bias[0] : 0.f);
}

// S[b,i,j] = masked softmax_j leaky_relu(s1[b,i] + s2[b,j] + attb, 0.2)
__global__ __launch_bounds__(128) void softmaxS_kernel(
    const float* __restrict__ s1, const float* __restrict__ s2,
    const float* __restrict__ attb, const unsigned char* __restrict__ mask,
    float* __restrict__ S) {
  __shared__ float red[128];
  int bi = blockIdx.x;
  int b = bi >> 7, i = bi & 127;
  int j = threadIdx.x;
  bool m = (j == i) || mask[((size_t)b * 128 + i) * 128 + j];
  float l = s1[bi] + s2[(b << 7) + j] + attb[0];
  l = l >= 0.f ? l : 0.2f * l;
  float z = m ? l : NEG_INF;
  red[j] = z;
  __syncthreads();
  for (int o = 64; o; o >>= 1) {
    if (j < o) red[j] = fmaxf(red[j], red[j + o]);
    __syncthreads();
  }
  float mx = red[0];
  __syncthreads();
  float e = m ? expf(l - mx) : 0.f;
  red[j] = e;
  __syncthreads();
  for (int o = 64; o; o >>= 1) {
    if (j < o) red[j] += red[j + o];
    __syncthreads();
  }
  float inv = 1.f / fmaxf(red[0], 1e-16f);
  S[((size_t)b * 128 + i) * 128 + j] = e * inv;
}

// LEConv fitness: fit = sigmoid(sum_j Af[i,j] amsg[j] - deg_i * bmsg_i + l3_i)
__global__ void fitness_kernel(const unsigned char* __restrict__ mask,
                               const float* __restrict__ amsg,
                               const float* __restrict__ bmsg,
                               const float* __restrict__ l3,
                               float* __restrict__ fit, int Bn) {
  int bi = blockIdx.x * blockDim.x + threadIdx.x;
  if (bi >= Bn) return;
  int b = bi >> 7, i = bi & 127;
  const unsigned char* mrow = mask + ((size_t)b * 128 + i) * 128;
  float acc = 0.f, deg = 0.f;
  for (int j = 0; j < 128; ++j) {
    bool m = (j == i) || mrow[j];
    if (m) {
      acc += amsg[(b << 7) + j];
      deg += 1.f;
    }
  }
  float x = acc - deg * bmsg[bi] + l3[bi];
  fit[bi] = 1.f / (1.f + expf(-x));
}

// top-64 (jax.lax.top_k semantics) + gather xs[b,p,:] = xn[b,idx_p,:]*vals_p
__global__ __launch_bounds__(128) void topk1_kernel(
    const float* __restrict__ fit, const float* __restrict__ xn,
    int* __restrict__ idx, float* __restrict__ vals, float* __restrict__ xs) {
  __shared__ float f[128];
  __shared__ int sidx[64];
  __shared__ float svals[64];
  int b = blockIdx.x, t = threadIdx.x;
  f[t] = fit[(b << 7) + t];
  __syncthreads();
  float mine = f[t];
  int rank = 0;
  for (int j = 0; j < 128; ++j) {
    float fj = f[j];
    rank += (fj > mine) || (fj == mine && j < t);
  }
  if (rank < 64) {
    sidx[rank] = t;
    svals[rank] = mine;
  }
  __syncthreads();
  if (t < 64) {
    idx[b * 64 + t] = sidx[t];
    vals[b * 64 + t] = svals[t];
  }
  for (int e = t; e < 64 * 384; e += 128) {
    int p = e / 384, c = e % 384;
    xs[((size_t)b * 64 + p) * 384 + c] =
        xn[((size_t)b * 128 + sidx[p]) * 384 + c] * svals[p];
  }
}

// mask2 = (Ssel Af Ssel^T > 0) | I  with Af = mask1 | I
__global__ __launch_bounds__(256) void coarsen1_kernel(
    const float* __restrict__ S, const unsigned char* __restrict__ mask1,
    const int* __restrict__ idx, unsigned char* __restrict__ mask2) {
  __shared__ float Sel[64][128];
  __shared__ float T[64][128];
  int b = blockIdx.x, t = threadIdx.x;
  const float* Sb = S + (size_t)b * 128 * 128;
  const unsigned char* Mb = mask1 + (size_t)b * 128 * 128;
  for (int e = t; e < 64 * 128; e += 256) {
    int p = e >> 7, u = e & 127;
    Sel[p][u] = Sb[(size_t)idx[b * 64 + p] * 128 + u];
  }
  __syncthreads();
  for (int e = t; e < 64 * 128; e += 256) {
    int p = e >> 7, vc = e & 127;
    float s = 0.f;
    for (int u = 0; u < 128; ++u) {
      bool m = (u == vc) || Mb[(size_t)u * 128 + vc];
      if (m) s += Sel[p][u];
    }
    T[p][vc] = s;
  }
  __syncthreads();
  for (int e = t; e < 64 * 64; e += 256) {
    int p = e >> 6, qn = e & 63;
    float s = 0.f;
    for (int vv = 0; vv < 128; ++vv) s += T[p][vv] * Sel[qn][vv];
    mask2[((size_t)b * 64 + p) * 64 + qn] = ((s > 0.f) || (p == qn)) ? 1 : 0;
  }
}

// stage-2 q/k/v/skip: x2[32768,384] @ W[384,2] + b, four weights at once
__global__ void t2lin_kernel(
    const float* __restrict__ x2, const float* __restrict__ Wq,
    const float* __restrict__ Wk, const float* __restrict__ Wv,
    const float* __restrict__ Ws, const float* __restrict__ bq,
    const float* __restrict__ bk, const float* __restrict__ bv,
    const float* __restrict__ bs, float* __restrict__ q2,
    float* __restrict__ k2, float* __restrict__ v2, float* __restrict__ x3,
    int Mrows) {
  int r = blockIdx.x * blockDim.x + threadIdx.x;
  if (r >= Mrows) return;
  const float* xr = x2 + (size_t)r * 384;
  float aq0 = 0, aq1 = 0, ak0 = 0, ak1 = 0, av0 = 0, av1 = 0, as0 = 0, as1 = 0;
  for (int c = 0; c < 384; ++c) {
    float xv = xr[c];
    aq0 += xv * Wq[c * 2];
    aq1 += xv * Wq[c * 2 + 1];
    ak0 += xv * Wk[c * 2];
    ak1 += xv * Wk[c * 2 + 1];
    av0 += xv * Wv[c * 2];
    av1 += xv * Wv[c * 2 + 1];
    as0 += xv * Ws[c * 2];
    as1 += xv * Ws[c * 2 + 1];
  }
  q2[r * 2] = aq0 + bq[0];
  q2[r * 2 + 1] = aq1 + bq[1];
  k2[r * 2] = ak0 + bk[0];
  k2[r * 2 + 1] = ak1 + bk[1];
  v2[r * 2] = av0 + bv[0];
  v2[r * 2 + 1] = av1 + bv[1];
  x3[r * 2] = as0 + bs[0];
  x3[r * 2 + 1] = as1 + bs[1];
}

// stage-2 attention: H=2, d=1, n=64; one block per graph
__global__ __launch_bounds__(64) void attn2_kernel(
    const float* __restrict__ q2, const float* __restrict__ k2,
    const float* __restrict__ v2, const unsigned char* __restrict__ mask2,
    float* __restrict__ x3) {
  __shared__ float kv[64][4];
  int b = blockIdx.x, i = threadIdx.x;
  kv[i][0] = k2[(b * 64 + i) * 2];
  kv[i][1] = k2[(b * 64 + i) * 2 + 1];
  kv[i][2] = v2[(b * 64 + i) * 2];
  kv[i][3] = v2[(b * 64 + i) * 2 + 1];
  __syncthreads();
  const unsigned char* mr = mask2 + ((size_t)b * 64 + i) * 64;
  for (int h = 0; h < 2; ++h) {
    float qi = q2[(b * 64 + i) * 2 + h];
    float mx = NEG_INF;
    for (int j = 0; j < 64; ++j)
      if (mr[j]) mx = fmaxf(mx, qi * kv[j][h]);
    float ssum = 0.f, o = 0.f;
    for (int j = 0; j < 64; ++j) {
      if (mr[j]) {
        float e = expf(qi * kv[j][h] - mx);
        ssum += e;
        o += e * kv[j][2 + h];
      }
    }
    o /= fmaxf(ssum, 1e-16f);
    x3[((size_t)b * 64 + i) * 2 + h] += o;
  }
}

// ASAP stage 2 fully fused (n=64, C=2, k=32) + global mean pool -> g[b,2]
__global__ __launch_bounds__(64) void asap2_kernel(
    const float* __restrict__ x3, const unsigned char* __restrict__ mask2,
    const float* __restrict__ linW, const float* __restrict__ linb,
    const float* __restrict__ attw, const float* __restrict__ attb,
    const float* __restrict__ le1W, const float* __restrict__ le1b,
    const float* __restrict__ le2W, const float* __restrict__ le3W,
    const float* __restrict__ le3b, float* __restrict__ g) {
  __shared__ float X[64][2];
  __shared__ float s2sh[64];
  __shared__ float xnsh[64][2];
  __shared__ float am[64], bm[64], l3s[64], fitv[64];
  __shared__ int sidx[32];
  __shared__ float svals[32];
  __shared__ float red[64];
  int b = blockIdx.x, i = threadIdx.x;
  const unsigned char* mr = mask2 + ((size_t)b * 64 + i) * 64;
  X[i][0] = x3[(b * 64 + i) * 2];
  X[i][1] = x3[(b * 64 + i) * 2 + 1];
  __syncthreads();
  float m0 = NEG_INF, m1 = NEG_INF;
  for (int j = 0; j < 64; ++j) {
    if ((j == i) || mr[j]) {
      m0 = fmaxf(m0, X[j][0]);
      m1 = fmaxf(m1, X[j][1]);
    }
  }
  float xq0 = m0 * linW[0] + m1 * linW[2] + linb[0];
  float xq1 = m0 * linW[1] + m1 * linW[3] + linb[1];
  float s1 = xq0 * attw[0] + xq1 * attw[1];
  s2sh[i] = X[i][0] * attw[2] + X[i][1] * attw[3];
  __syncthreads();
  float mx = NEG_INF;
  for (int j = 0; j < 64; ++j) {
    if ((j == i) || mr[j]) {
      float l = s1 + s2sh[j] + attb[0];
      l = l >= 0.f ? l : 0.2f * l;
      mx = fmaxf(mx, l);
    }
  }
  float ssum = 0.f, xn0 = 0.f, xn1 = 0.f;
  for (int j = 0; j < 64; ++j) {
    if ((j == i) || mr[j]) {
      float l = s1 + s2sh[j] + attb[0];
      l = l >= 0.f ? l : 0.2f * l;
      float e = expf(l - mx);
      ssum += e;
      xn0 += e * X[j][0];
      xn1 += e * X[j][1];
    }
  }
  float inv = 1.f / fmaxf(ssum, 1e-16f);
  xn0 *= inv;
  xn1 *= inv;
  xnsh[i][0] = xn0;
  xnsh[i][1] = xn1;
  am[i] = xn0 * le1W[0] + xn1 * le1W[1] + le1b[0];
  bm[i] = xn0 * le2W[0] + xn1 * le2W[1];
  l3s[i] = xn0 * le3W[0] + xn1 * le3W[1] + le3b[0];
  __syncthreads();
  float acc = 0.f, deg = 0.f;
  for (int j = 0; j < 64; ++j) {
    if ((j == i) || mr[j]) {
      acc += am[j];
      deg += 1.f;
    }
  }
  float fx = acc - deg * bm[i] + l3s[i];
  fitv[i] = 1.f / (1.f + expf(-fx));
  __syncthreads();
  float mine = fitv[i];
  int rank = 0;
  for (int j = 0; j < 64; ++j)
    rank += (fitv[j] > mine) || (fitv[j] == mine && j < i);
  if (rank < 32) {
    sidx[rank] = i;
    svals[rank] = mine;
  }
  __syncthreads();
  float c0 = 0.f, c1 = 0.f;
  if (i < 32) {
    c0 = xnsh[sidx[i]][0] * svals[i];
    c1 = xnsh[sidx[i]][1] * svals[i];
  }
  red[i] = c0;
  __syncthreads();
  for (int o = 32; o; o >>= 1) {
    if (i < o) red[i] += red[i + o];
    __syncthreads();
  }
  if (i == 0) g[b * 2] = red[0] / 32.f;
  __syncthreads();
  red[i] = c1;
  __syncthreads();
  for (int o = 32; o; o >>= 1) {
    if (i < o) red[i] += red[i + o];
    __syncthreads();
  }
  if (i == 0) g[b * 2 + 1] = red[0] / 32.f;
}

// Collapse the two linear MLP heads (no activation between layers).
__global__ void headpre_kernel(
    const float* __restrict__ oW1, const float* __restrict__ ob1,
    const float* __restrict__ oW2, const float* __restrict__ ob2,
    const float* __restrict__ bW1, const float* __restrict__ bb1,
    const float* __restrict__ bW2, const float* __restrict__ bb2,
    float* __restrict__ weff, float* __restrict__ beff,
    float* __restrict__ veff, float* __restrict__ ceff) {
  int t = threadIdx.x;
  if (t < 32) {
    float s = 0.f;
    for (int h = 0; h < 128; ++h) s += oW1[t * 128 + h] * oW2[h];
    weff[t] = s;
  } else if (t == 32) {
    float s = 0.f;
    for (int h = 0; h < 128; ++h) s += ob1[h] * oW2[h];
    beff[0] = s + ob2[0];
  } else if (t >= 40 && t < 45) {
    int j = t - 40;
    float s = 0.f;
    for (int h = 0; h < 128; ++h) s += bW1[j * 128 + h] * bW2[h];
    veff[j] = s;
  } else if (t == 45) {
    float s = 0.f;
    for (int h = 0; h < 128; ++h) s += bb1[h] * bW2[h];
    ceff[0] = s + bb2[0];
  }
}

__global__ void final_kernel(
    const float* __restrict__ g, const float* __restrict__ noisy,
    const float* __restrict__ obs, const float* __restrict__ depth,
    const float* __restrict__ weff, const float* __restrict__ beff,
    const float* __restrict__ veff, const float* __restrict__ ceff,
    float* __restrict__ out, int Bq) {
  int b = blockIdx.x * blockDim.x + threadIdx.x;
  if (b >= Bq) return;
  float om = 0.f;
  for (int l = 0; l < 16; ++l) {
    float s = beff[0];
    const float* row = obs + ((size_t)b * 16 + l) * 32;
    for (int f = 0; f < 32; ++f) s += row[f] * weff[f];
    om += s;
  }
  om *= (1.f / 16.f);
  out[b] = g[b * 2] * veff[0] + g[b * 2 + 1] * veff[1] + noisy[b] * veff[2] +
           om * veff[3] + depth[b] * veff[4] + ceff[0];
}

// ---------------------------------------------------------------------------
extern "C" void kernel_launch(void* const* d_in, const int* in_sizes, int n_in,
                              void* d_out, int out_size, void* d_ws,
                              size_t ws_size, hipStream_t stream) {
  (void)n_in;
  (void)out_size;
  (void)ws_size;
  const int Bg = 512;
  const size_t M1 = (size_t)Bg * 128;  // 65536 nodes stage 1
  const int C1 = 384;

  const float* Xn = (const float*)d_in[0];
  const int* edges = (const int*)d_in[1];
  const float* noisy = (const float*)d_in[2];
  const float* obs = (const float*)d_in[3];
  const float* depth = (const float*)d_in[4];
  const float* t1_Wq = (const float*)d_in[6];
  const float* t1_Wk = (const float*)d_in[7];
  const float* t1_Wv = (const float*)d_in[8];
  const float* t1_Ws = (const float*)d_in[9];
  const float* t1_bq = (const float*)d_in[10];
  const float* t1_bk = (const float*)d_in[11];
  const float* t1_bv = (const float*)d_in[12];
  const float* t1_bs = (const float*)d_in[13];
  const float* p1_linW = (const float*)d_in[14];
  const float* p1_linb = (const float*)d_in[15];
  const float* p1_attw = (const float*)d_in[16];
  const float* p1_attb = (const float*)d_in[17];
  const float* p1_le1W = (const float*)d_in[18];
  const float* p1_le1b = (const float*)d_in[19];
  const float* p1_le2W = (const float*)d_in[20];
  const float* p1_le3W = (const float*)d_in[21];
  const float* p1_le3b = (const float*)d_in[22];
  const float* t2_Wq = (const float*)d_in[23];
  const float* t2_Wk = (const float*)d_in[24];
  const float* t2_Wv = (const float*)d_in[25];
  const float* t2_Ws = (const float*)d_in[26];
  const float* t2_bq = (const float*)d_in[27];
  const float* t2_bk = (const float*)d_in[28];
  const float* t2_bv = (const float*)d_in[29];
  const float* t2_bs = (const float*)d_in[30];
  const float* p2_linW = (const float*)d_in[31];
  const float* p2_linb = (const float*)d_in[32];
  const float* p2_attw = (const float*)d_in[33];
  const float* p2_attb = (const float*)d_in[34];
  const float* p2_le1W = (const float*)d_in[35];
  const float* p2_le1b = (const float*)d_in[36];
  const float* p2_le2W = (const float*)d_in[37];
  const float* p2_le3W = (const float*)d_in[38];
  const float* p2_le3b = (const float*)d_in[39];
  const float* obs_W1 = (const float*)d_in[40];
  const float* obs_b1 = (const float*)d_in[41];
  const float* obs_W2 = (const float*)d_in[42];
  const float* obs_b2 = (const float*)d_in[43];
  const float* body_W1 = (const float*)d_in[44];
  const float* body_b1 = (const float*)d_in[45];
  const float* body_W2 = (const float*)d_in[46];
  const float* body_b2 = (const float*)d_in[47];

  char* ws = (char*)d_ws;
  size_t off = 0;
  auto alloc = [&](size_t bytes) -> void* {
    void* p = ws + off;
    off += (bytes + 255) & ~(size_t)255;
    return p;
  };
  unsigned char* mask1 = (unsigned char*)alloc((size_t)Bg * 128 * 128);
  unsigned char* mask2 = (unsigned char*)alloc((size_t)Bg * 64 * 64);
  float* bufQ = (float*)alloc(M1 * C1 * sizeof(float));  // q1 -> xqmax -> xn1
  float* bufK = (float*)alloc(M1 * C1 * sizeof(float));  // k1 -> xq_lin -> x2
  float* bufV = (float*)alloc(M1 * C1 * sizeof(float));  // v1 -> S1
  float* x1 = (float*)alloc(M1 * C1 * sizeof(float));
  float* s1v = (float*)alloc(M1 * sizeof(float));
  float* s2v = (float*)alloc(M1 * sizeof(float));
  float* amsg = (float*)alloc(M1 * sizeof(float));
  float* bmsg = (float*)alloc(M1 * sizeof(float));
  float* l3v = (float*)alloc(M1 * sizeof(float));
  float* fitv = (float*)alloc(M1 * sizeof(float));
  int* idx1 = (int*)alloc((size_t)Bg * 64 * sizeof(int));
  float* vals1 = (float*)alloc((size_t)Bg * 64 * sizeof(float));
  float* q2 = (float*)alloc((size_t)Bg * 64 * 2 * sizeof(float));
  float* k2 = (float*)alloc((size_t)Bg * 64 * 2 * sizeof(float));
  float* v2 = (float*)alloc((size_t)Bg * 64 * 2 * sizeof(float));
  float* x3 = (float*)alloc((size_t)Bg * 64 * 2 * sizeof(float));
  float* gbuf = (float*)alloc((size_t)Bg * 2 * sizeof(float));
  float* weff = (float*)alloc(32 * sizeof(float));
  float* beff = (float*)alloc(sizeof(float));
  float* veff = (float*)alloc(8 * sizeof(float));
  float* ceff = (float*)alloc(sizeof(float));

  // 1. adjacency mask
  size_t mb = (size_t)Bg * 128 * 128;
  zero_u8_kernel<<<(int)((mb + 255) / 256), 256, 0, stream>>>(mask1, mb);
  int E = in_sizes[1] / 2;
  edges_kernel<<<(E + 255) / 256, 256, 0, stream>>>(edges, mask1, E);

  // 2. q/k/v + root skip (x1)   [65536x64]@[64x384]
  {
    dim3 grid((unsigned)(M1 / 64), C1 / 16);
    gemm_bias_kernel<<<grid, 128, 0, stream>>>(Xn, t1_Wq, t1_bq, bufQ, (int)M1, C1, 64);
    gemm_bias_kernel<<<grid, 128, 0, stream>>>(Xn, t1_Wk, t1_bk, bufK, (int)M1, C1, 64);
    gemm_bias_kernel<<<grid, 128, 0, stream>>>(Xn, t1_Wv, t1_bv, bufV, (int)M1, C1, 64);
    gemm_bias_kernel<<<grid, 128, 0, stream>>>(Xn, t1_Ws, t1_bs, x1, (int)M1, C1, 64);
  }
  // 3. attention (3 heads x 512 graphs)
  attn1_kernel<<<dim3(3, Bg), 256, 0, stream>>>(bufQ, bufK, bufV, mask1, x1);

  // 4. ASAP1: scatter-max -> lin -> att scores -> S -> xn
  scatmax_kernel<<<(int)M1, 128, 0, stream>>>(x1, mask1, bufQ);
  {
    dim3 grid((unsigned)(M1 / 64), C1 / 16);
    gemm_bias_kernel<<<grid, 128, 0, stream>>>(bufQ, p1_linW, p1_linb, bufK, (int)M1, C1, C1);
  }
  matvec_kernel<<<(int)(M1 / 8), 256, 0, stream>>>(bufK, p1_attw, nullptr, s1v, (int)M1, C1);
  matvec_kernel<<<(int)(M1 / 8), 256, 0, stream>>>(x1, p1_attw + C1, nullptr, s2v, (int)M1, C1);
  softmaxS_kernel<<<(int)M1, 128, 0, stream>>>(s1v, s2v, p1_attb, mask1, bufV);
  gemm_batched_kernel<<<dim3(C1 / 16, Bg), 128, 0, stream>>>(
      bufV, x1, bufQ, 128, C1, 128, (size_t)128 * 128, (size_t)128 * C1, (size_t)128 * C1);

  // 5. LEConv fitness + top-64 + gather + coarsened mask
  matvec_kernel<<<(int)(M1 / 8), 256, 0, stream>>>(bufQ, p1_le1W, p1_le1b, amsg, (int)M1, C1);
  matvec_kernel<<<(int)(M1 / 8), 256, 0, stream>>>(bufQ, p1_le2W, nullptr, bmsg, (int)M1, C1);
  matvec_kernel<<<(int)(M1 / 8), 256, 0, stream>>>(bufQ, p1_le3W, p1_le3b, l3v, (int)M1, C1);
  fitness_kernel<<<(int)(M1 / 256), 256, 0, stream>>>(mask1, amsg, bmsg, l3v, fitv, (int)M1);
  topk1_kernel<<<Bg, 128, 0, stream>>>(fitv, bufQ, idx1, vals1, bufK);
  coarsen1_kernel<<<Bg, 256, 0, stream>>>(bufV, mask1, idx1, mask2);

  // 6. stage-2 transformer conv + ASAP2 + mean pool
  t2lin_kernel<<<(Bg * 64 + 255) / 256, 256, 0, stream>>>(
      bufK, t2_Wq, t2_Wk, t2_Wv, t2_Ws, t2_bq, t2_bk, t2_bv, t2_bs, q2, k2, v2, x3, Bg * 64);
  attn2_kernel<<<Bg, 64, 0, stream>>>(q2, k2, v2, mask2, x3);
  asap2_kernel<<<Bg, 64, 0, stream>>>(x3, mask2, p2_linW, p2_linb, p2_attw, p2_attb,
                                      p2_le1W, p2_le1b, p2_le2W, p2_le3W, p2_le3b, gbuf);

  // 7. collapsed linear heads + final merge
  headpre_kernel<<<1, 64, 0, stream>>>(obs_W1, obs_b1, obs_W2, obs_b2,
                                       body_W1, body_b1, body_W2, body_b2,
                                       weff, beff, veff, ceff);
  final_kernel<<<(Bg + 255) / 256, 256, 0, stream>>>(gbuf, noisy, obs, depth, weff, beff,
                                                     veff, ceff, (float*)d_out, Bg);
}